// GCNNet_40535901339688
// MI455X (gfx1250) — compile-verified
//
#include <hip/hip_runtime.h>
#include <math.h>

#define N_NODES 100000
#define N_EDGES 3200000
#define BN_EPS 1e-5f

typedef __attribute__((ext_vector_type(2))) float v2f;
typedef __attribute__((ext_vector_type(8))) float v8f;

// ---------------- degree / normalization ----------------
__global__ void k_init_deg(float* deg, int n) {
    int i = blockIdx.x * blockDim.x + threadIdx.x;
    if (i < n) deg[i] = 1.0f;                    // self-loop
}

__global__ void k_count_deg(const int* __restrict__ dst, float* deg, int e) {
    int i = blockIdx.x * blockDim.x + threadIdx.x;
    if (i < e) atomicAdd(&deg[dst[i]], 1.0f);
}

__global__ void k_rsqrt_inplace(float* deg, int n) {
    int i = blockIdx.x * blockDim.x + threadIdx.x;
    if (i < n) deg[i] = rsqrtf(deg[i]);          // dis = deg^-1/2
}

// ---------------- WMMA f32 GEMM: T[N,dout] = X[N,K] @ W[K,dout] ----------------
// One wave computes a 16x16 output tile with V_WMMA_F32_16X16X4_F32 over K/4 steps.
// A (16x4 f32) per ISA layout: lanes 0-15 M=0..15 half0, lanes 16-31 half1;
//   VGPR0 holds K = 2*half, VGPR1 holds K = 2*half+1.
// B (4x16 f32): VGPR0 row K=half, VGPR1 row K=half+2, N = lane&15.
// C/D (16x16 f32, 8 VGPRs): VGPR r -> M = r + 8*half, N = lane&15.
__global__ void k_gemm_wmma(const float* __restrict__ X, const float* __restrict__ W,
                            float* __restrict__ T, int K, int dout,
                            int col_tiles, int row_tiles) {
    int wid  = (blockIdx.x * blockDim.x + threadIdx.x) >> 5;  // wave id (uniform)
    int lane = threadIdx.x & 31;
    int rt = wid / col_tiles;
    int ct = wid - rt * col_tiles;
    if (rt >= row_tiles) return;                 // whole-wave uniform exit

    int half = lane >> 4;                        // 0 | 1
    int row  = rt * 16 + (lane & 15);
    int col  = ct * 16 + (lane & 15);

    const float* xrow = X + (size_t)row * K + 2 * half;   // A K-offset per half
    const float* wcol = W + (size_t)half * dout + col;    // B K-row per half

    v8f acc = {};
    for (int k = 0; k < K; k += 4) {
        v2f a, b;
        a[0] = xrow[k];
        a[1] = xrow[k + 1];
        b[0] = wcol[(size_t)k * dout];
        b[1] = wcol[(size_t)(k + 2) * dout];
        acc = __builtin_amdgcn_wmma_f32_16x16x4_f32(
            /*neg_a=*/false, a, /*neg_b=*/false, b,
            /*c_mod=*/(short)0, acc, /*reuse_a=*/false, /*reuse_b=*/false);
    }

    float* trow = T + (size_t)(rt * 16 + 8 * half) * dout + col;
#pragma unroll
    for (int r = 0; r < 8; ++r)
        trow[(size_t)r * dout] = acc[r];
}

// ---------------- self-loop + bias init: AGG = T*dis^2 + b ----------------
__global__ void k_self_bias(const float* __restrict__ T, const float* __restrict__ dis,
                            const float* __restrict__ b, float* __restrict__ AGG,
                            int n, int dout) {
    int i = blockIdx.x * blockDim.x + threadIdx.x;
    if (i >= n * dout) return;
    int node = i / dout;
    int f    = i - node * dout;
    float ds = dis[node];
    AGG[i] = T[i] * ds * ds + b[f];
}

// ---------------- edge scatter: AGG[dst] += T[src] * dis[src]*dis[dst] ----------------
// Wave per edge; lanes stride the feature dim -> coalesced loads & atomics.
__global__ void k_scatter(const float* __restrict__ T, const int* __restrict__ src,
                          const int* __restrict__ dst, const float* __restrict__ dis,
                          float* __restrict__ AGG, int e, int dout) {
    int wid  = (blockIdx.x * blockDim.x + threadIdx.x) >> 5;
    int lane = threadIdx.x & 31;
    if (wid >= e) return;
    int s = src[wid], d = dst[wid];
    float c = dis[s] * dis[d];
    const float* ts = T + (size_t)s * dout;
    float* ad = AGG + (size_t)d * dout;
    for (int f = lane; f < dout; f += 32)
        atomicAdd(&ad[f], ts[f] * c);
}

// ---------------- BN + ELU (in place) ----------------
__device__ __forceinline__ float elu1(float x) {
    return x > 0.0f ? x : (__expf(x) - 1.0f);
}

__global__ void k_bn_elu(float* __restrict__ H, const float* __restrict__ g,
                         const float* __restrict__ be, const float* __restrict__ m,
                         const float* __restrict__ v, int n, int dout) {
    int i = blockIdx.x * blockDim.x + threadIdx.x;
    if (i >= n * dout) return;
    int f = i & (dout - 1);                      // dout is a power of two (32/64/128)
    float x = H[i];
    x = g[f] * (x - m[f]) * rsqrtf(v[f] + BN_EPS) + be[f];
    H[i] = elu1(x);
}

// ---------------- layer 4 GEMV (128 -> 1): wave per node, shuffle reduce ----------------
__global__ void k_gemv128(const float* __restrict__ X, const float* __restrict__ W,
                          float* __restrict__ T, int n) {
    int wid  = (blockIdx.x * blockDim.x + threadIdx.x) >> 5;
    int lane = threadIdx.x & 31;
    if (wid >= n) return;
    const float* xr = X + (size_t)wid * 128;
    float s = 0.0f;
#pragma unroll
    for (int k = 0; k < 128; k += 32)
        s += xr[k + lane] * W[k + lane];
#pragma unroll
    for (int off = 16; off > 0; off >>= 1)
        s += __shfl_down(s, off, 32);
    if (lane == 0) T[wid] = s;
}

// ---------------- final sigmoid(elu(x)) in place on d_out ----------------
__global__ void k_sig_elu(float* __restrict__ O, int n) {
    int i = blockIdx.x * blockDim.x + threadIdx.x;
    if (i >= n) return;
    float x = elu1(O[i]);
    O[i] = 1.0f / (1.0f + __expf(-x));
}

// =====================================================================
extern "C" void kernel_launch(void* const* d_in, const int* in_sizes, int n_in,
                              void* d_out, int out_size, void* d_ws, size_t ws_size,
                              hipStream_t stream) {
    (void)in_sizes; (void)n_in; (void)out_size; (void)ws_size;

    const float* x   = (const float*)d_in[0];
    const int*   src = (const int*)d_in[1];
    const int*   dst = (const int*)d_in[2];
    const float* W1  = (const float*)d_in[3];   const float* b1  = (const float*)d_in[4];
    const float* W2  = (const float*)d_in[5];   const float* b2  = (const float*)d_in[6];
    const float* W3  = (const float*)d_in[7];   const float* b3  = (const float*)d_in[8];
    const float* W4  = (const float*)d_in[9];   const float* b4  = (const float*)d_in[10];
    const float* g1  = (const float*)d_in[11];  const float* be1 = (const float*)d_in[12];
    const float* m1  = (const float*)d_in[13];  const float* v1  = (const float*)d_in[14];
    const float* g2  = (const float*)d_in[15];  const float* be2 = (const float*)d_in[16];
    const float* m2  = (const float*)d_in[17];  const float* v2  = (const float*)d_in[18];
    const float* g3  = (const float*)d_in[19];  const float* be3 = (const float*)d_in[20];
    const float* m3  = (const float*)d_in[21];  const float* v3  = (const float*)d_in[22];

    // workspace layout (floats): dis[N] | bufT[N*128] | bufA[N*128] | bufB[N*64]
    float* dis  = (float*)d_ws;
    float* bufT = dis  + N_NODES;
    float* bufA = bufT + (size_t)N_NODES * 128;
    float* bufB = bufA + (size_t)N_NODES * 128;
    float* out  = (float*)d_out;

    const int TB = 256;                              // 8 wave32s per block
    const int row_tiles = N_NODES / 16;              // 6250 (exact)

    auto cdiv = [](long a, long b) { return (unsigned)((a + b - 1) / b); };

    // --- normalization coefficients ---
    k_init_deg <<<cdiv(N_NODES, TB), TB, 0, stream>>>(dis, N_NODES);
    k_count_deg<<<cdiv(N_EDGES, TB), TB, 0, stream>>>(dst, dis, N_EDGES);
    k_rsqrt_inplace<<<cdiv(N_NODES, TB), TB, 0, stream>>>(dis, N_NODES);

    auto conv = [&](const float* Hin, const float* W, const float* b,
                    float* AGG, int K, int dout) {
        int ct = dout / 16;
        long waves = (long)row_tiles * ct;
        k_gemm_wmma<<<cdiv(waves, 8), TB, 0, stream>>>(Hin, W, bufT, K, dout,
                                                       ct, row_tiles);
        k_self_bias<<<cdiv((long)N_NODES * dout, TB), TB, 0, stream>>>(
            bufT, dis, b, AGG, N_NODES, dout);
        k_scatter<<<cdiv((long)N_EDGES, 8), TB, 0, stream>>>(
            bufT, src, dst, dis, AGG, N_EDGES, dout);
    };

    // layer 1: 128 -> 32
    conv(x, W1, b1, bufA, 128, 32);
    k_bn_elu<<<cdiv((long)N_NODES * 32, TB), TB, 0, stream>>>(bufA, g1, be1, m1, v1, N_NODES, 32);
    // layer 2: 32 -> 64
    conv(bufA, W2, b2, bufB, 32, 64);
    k_bn_elu<<<cdiv((long)N_NODES * 64, TB), TB, 0, stream>>>(bufB, g2, be2, m2, v2, N_NODES, 64);
    // layer 3: 64 -> 128
    conv(bufB, W3, b3, bufA, 64, 128);
    k_bn_elu<<<cdiv((long)N_NODES * 128, TB), TB, 0, stream>>>(bufA, g3, be3, m3, v3, N_NODES, 128);

    // layer 4: 128 -> 1 (GEMV + scatter directly into d_out)
    k_gemv128<<<cdiv((long)N_NODES * 32, TB), TB, 0, stream>>>(bufA, W4, bufT, N_NODES);
    k_self_bias<<<cdiv(N_NODES, TB), TB, 0, stream>>>(bufT, dis, b4, out, N_NODES, 1);
    k_scatter<<<cdiv((long)N_EDGES, 8), TB, 0, stream>>>(bufT, src, dst, dis, out, N_EDGES, 1);
    k_sig_elu<<<cdiv(N_NODES, TB), TB, 0, stream>>>(out, N_NODES);
}